// Sinkhorn_13709535609454
// MI455X (gfx1250) — compile-verified
//
#include <hip/hip_runtime.h>
#include <cstdint>
#include <cstddef>

// Batched log-domain Sinkhorn, dual-potential formulation.
// s: (256, 512, 512) fp32.  out = exp(s/TAU - u_i - v_j) after 10 alternating
// row/col logsumexp updates of potentials u, v (512 floats each, in LDS).

#define TAU_INV   20.0f          // 1 / 0.05
#define NN        512
#define THREADS   512            // 16 waves of 32
#define PROWS     32             // panel rows streamed through LDS (col phases)
#define PSTRIDE   520            // padded row stride (floats): bank shift 8/row
#define NPANELS   (NN / PROWS)   // 16

typedef float v2f __attribute__((ext_vector_type(2)));
typedef float v8f __attribute__((ext_vector_type(8)));
typedef int   v4i __attribute__((ext_vector_type(4)));

#if defined(__AMDGCN__) && __has_builtin(__builtin_amdgcn_wmma_f32_16x16x4_f32)
#define HAVE_WMMA 1
#else
#define HAVE_WMMA 0
#endif

#if defined(__AMDGCN__) && __has_builtin(__builtin_amdgcn_global_load_async_to_lds_b128) && \
    __has_builtin(__builtin_amdgcn_s_wait_asynccnt)
#define HAVE_ASYNC 1
#else
#define HAVE_ASYNC 0
#endif

#if HAVE_ASYNC
#define ASYNC_WAIT0() __builtin_amdgcn_s_wait_asynccnt(0)
typedef __attribute__((address_space(1))) v4i* g_v4i_p;   // global v4i*
typedef __attribute__((address_space(3))) v4i* l_v4i_p;   // LDS v4i*
#else
#define ASYNC_WAIT0() ((void)0)
#endif

__device__ __forceinline__ float wave_sum(float x) {
#pragma unroll
  for (int off = 16; off >= 1; off >>= 1) x += __shfl_xor(x, off, 32);
  return x;
}

__device__ __forceinline__ float wave_max(float x) {
#pragma unroll
  for (int off = 16; off >= 1; off >>= 1) x = fmaxf(x, __shfl_xor(x, off, 32));
  return x;
}

// Copy a PROWS x 512 fp32 panel (row-major) into LDS with padded stride.
__device__ __forceinline__ void stage_panel(const float* src, float* buf, int tid) {
#if HAVE_ASYNC
#pragma unroll
  for (int j = 0; j < (PROWS * NN / 4) / THREADS; ++j) {   // 8 iters of b128
    int idx = tid + j * THREADS;
    int r  = idx >> 7;      // 128 float4 per row
    int c4 = idx & 127;
    __attribute__((address_space(1))) void* gp =
        (__attribute__((address_space(1))) void*)(src + r * NN + c4 * 4);
    __attribute__((address_space(3))) void* lp =
        (__attribute__((address_space(3))) void*)(buf + r * PSTRIDE + c4 * 4);
    __builtin_amdgcn_global_load_async_to_lds_b128((g_v4i_p)gp, (l_v4i_p)lp, 0, 0);
  }
#else
#pragma unroll
  for (int j = 0; j < (PROWS * NN / 4) / THREADS; ++j) {
    int idx = tid + j * THREADS;
    int r  = idx >> 7;
    int c4 = idx & 127;
    float4 t = *(const float4*)(src + r * NN + c4 * 4);
    *(float4*)(buf + r * PSTRIDE + c4 * 4) = t;
  }
#endif
}

// Row phase (iters 2,4,6,8): u[r] += log sum_j exp(s/tau - u[r] - v[j]).
// All entries <= 0 after previous normalization -> no max pass needed.
__device__ __forceinline__ void row_phase(const float* S, float* u, const float* v,
                                          int wave, int lane) {
  const float4* v4 = (const float4*)v;
#pragma unroll 1
  for (int r = wave * 32; r < wave * 32 + 32; ++r) {
    const float4* row = (const float4*)(S + (size_t)r * NN);
    float ur  = u[r];
    float sum = 0.f;
#pragma unroll
    for (int q = 0; q < 4; ++q) {
      int c4 = lane + 32 * q;
      float4 x = row[c4];
      float4 w = v4[c4];
      sum += __expf(x.x * TAU_INV - ur - w.x);
      sum += __expf(x.y * TAU_INV - ur - w.y);
      sum += __expf(x.z * TAU_INV - ur - w.z);
      sum += __expf(x.w * TAU_INV - ur - w.w);
    }
    sum = wave_sum(sum);
    if (lane == 0) u[r] = ur + __logf(sum);
  }
}

// Column phase (iters 1,3,5,7,9): v[c] += log sum_r exp(s/tau - u[r] - v[c]).
// Streams 32-row panels through double-buffered LDS (async DMA), reduces the
// row dimension with V_WMMA_F32_16X16X4_F32 using an all-ones A operand:
//   D = ones(16x4) x E(4x16) + C  -> column sums of E replicated over M.
__device__ __forceinline__ void col_phase(const float* S, const float* u, float* v,
                                          float* panel0, float* panel1,
                                          int tid, int wave, int lane) {
  const int   ln15 = lane & 15;
  const int   cA   = wave * 32 + ln15;   // this wave owns columns [wave*32, wave*32+32)
  const int   cB   = cA + 16;
  const float vcA  = v[cA];
  const float vcB  = v[cB];
  const int   q0   = (lane < 16) ? 0 : 2;   // half-wave -> K rows {0,1} vs {2,3}

  v8f acc0 = {0.f, 0.f, 0.f, 0.f, 0.f, 0.f, 0.f, 0.f};
  v8f acc1 = acc0;
#if HAVE_WMMA
  const v2f aones = {1.f, 1.f};
#endif

  stage_panel(S, panel0, tid);
  ASYNC_WAIT0();
  __syncthreads();

  for (int p = 0; p < NPANELS; ++p) {
    float* buf = (p & 1) ? panel1 : panel0;
    if (p + 1 < NPANELS)
      stage_panel(S + (size_t)(p + 1) * PROWS * NN, (p & 1) ? panel0 : panel1, tid);

#pragma unroll
    for (int ch = 0; ch < PROWS / 4; ++ch) {
      const int    kr0 = ch * 4 + q0;
      const float  u0  = u[p * PROWS + kr0];
      const float  u1  = u[p * PROWS + kr0 + 1];
      const float* r0  = buf + kr0 * PSTRIDE;
      const float* r1  = r0 + PSTRIDE;
      float e00 = __expf(r0[cA] * TAU_INV - u0 - vcA);
      float e01 = __expf(r1[cA] * TAU_INV - u1 - vcA);
      float e10 = __expf(r0[cB] * TAU_INV - u0 - vcB);
      float e11 = __expf(r1[cB] * TAU_INV - u1 - vcB);
#if HAVE_WMMA
      v2f b0 = {e00, e01};
      v2f b1 = {e10, e11};
      acc0 = __builtin_amdgcn_wmma_f32_16x16x4_f32(false, aones, false, b0,
                                                   (short)0, acc0, false, false);
      acc1 = __builtin_amdgcn_wmma_f32_16x16x4_f32(false, aones, false, b1,
                                                   (short)0, acc1, false, false);
#else
      acc0[0] += e00 + e01;
      acc1[0] += e10 + e11;
#endif
    }
    ASYNC_WAIT0();      // panel p+1 landed; all waves done with panel p
    __syncthreads();
  }

  float s0 = acc0[0];   // WMMA: lane n (n<16) holds full column sum for cA
  float s1 = acc1[0];
#if !HAVE_WMMA
  s0 += __shfl_xor(s0, 16, 32);   // combine half-wave partials
  s1 += __shfl_xor(s1, 16, 32);
#endif
  if (lane < 16) {
    v[cA] = vcA + __logf(s0);
    v[cB] = vcB + __logf(s1);
  }
  __syncthreads();
}

__global__ __launch_bounds__(THREADS)
void sinkhorn_kernel(const float* __restrict__ s, float* __restrict__ out) {
  __shared__ __align__(16) float u_s[NN];
  __shared__ __align__(16) float v_s[NN];
  __shared__ __align__(16) float panel[2][PROWS * PSTRIDE];

  const int tid  = threadIdx.x;
  const int lane = tid & 31;
  const int wave = tid >> 5;

  const float* S = s   + (size_t)blockIdx.x * NN * NN;
  float*       O = out + (size_t)blockIdx.x * NN * NN;

  u_s[tid] = 0.f;
  v_s[tid] = 0.f;
  __syncthreads();

  // ---- iter 0: row logsumexp with true max (raw s/tau spans +-~100) ----
#pragma unroll 1
  for (int r = wave * 32; r < wave * 32 + 32; ++r) {
    const float4* row = (const float4*)(S + (size_t)r * NN);
    float a[16];
#pragma unroll
    for (int q = 0; q < 4; ++q) {
      float4 t = row[lane + 32 * q];
      a[4 * q + 0] = t.x * TAU_INV;
      a[4 * q + 1] = t.y * TAU_INV;
      a[4 * q + 2] = t.z * TAU_INV;
      a[4 * q + 3] = t.w * TAU_INV;
    }
    float m = a[0];
#pragma unroll
    for (int k = 1; k < 16; ++k) m = fmaxf(m, a[k]);
    m = wave_max(m);
    float sum = 0.f;
#pragma unroll
    for (int k = 0; k < 16; ++k) sum += __expf(a[k] - m);
    sum = wave_sum(sum);
    if (lane == 0) u_s[r] = m + __logf(sum);
  }
  __syncthreads();

  // ---- iters 1..9: alternate col / row phases (no max needed) ----
  col_phase(S, u_s, v_s, panel[0], panel[1], tid, wave, lane);  // 1
  row_phase(S, u_s, v_s, wave, lane); __syncthreads();          // 2
  col_phase(S, u_s, v_s, panel[0], panel[1], tid, wave, lane);  // 3
  row_phase(S, u_s, v_s, wave, lane); __syncthreads();          // 4
  col_phase(S, u_s, v_s, panel[0], panel[1], tid, wave, lane);  // 5
  row_phase(S, u_s, v_s, wave, lane); __syncthreads();          // 6
  col_phase(S, u_s, v_s, panel[0], panel[1], tid, wave, lane);  // 7
  row_phase(S, u_s, v_s, wave, lane); __syncthreads();          // 8
  col_phase(S, u_s, v_s, panel[0], panel[1], tid, wave, lane);  // 9

  // ---- output: exp(s/tau - u_i - v_j), one read + one write pass ----
  {
    const float4* S4 = (const float4*)S;
    float4*       O4 = (float4*)O;
    const float4* v4 = (const float4*)v_s;
#pragma unroll 1
    for (int j = 0; j < (NN * NN / 4) / THREADS; ++j) {
      int idx = tid + j * THREADS;
      int r   = idx >> 7;
      int c4  = idx & 127;
      float4 x = S4[idx];
      float  ur = u_s[r];
      float4 w  = v4[c4];
      float4 o;
      o.x = __expf(x.x * TAU_INV - ur - w.x);
      o.y = __expf(x.y * TAU_INV - ur - w.y);
      o.z = __expf(x.z * TAU_INV - ur - w.z);
      o.w = __expf(x.w * TAU_INV - ur - w.w);
      O4[idx] = o;
    }
  }
}

extern "C" void kernel_launch(void* const* d_in, const int* in_sizes, int n_in,
                              void* d_out, int out_size, void* d_ws, size_t ws_size,
                              hipStream_t stream) {
  const float* s = (const float*)d_in[0];
  float* out = (float*)d_out;
  int nb = in_sizes[0] / (NN * NN);   // 256 batches, one workgroup per batch matrix
  sinkhorn_kernel<<<nb, THREADS, 0, stream>>>(s, out);
}